// RGCNLayer_55628416417805
// MI455X (gfx1250) — compile-verified
//
#include <hip/hip_runtime.h>
#include <hip/hip_bf16.h>
#include <float.h>

#define N_NODES 50000
#define N_FEAT  128
#define NUM_RELS 8

typedef __attribute__((ext_vector_type(2))) float v2f;
typedef __attribute__((ext_vector_type(8))) float v8f;

// ---------------------------------------------------------------------------
// Native CDNA5 float atomic max (GLOBAL_ATOMIC_MAX_NUM_F32, no return),
// device scope so it is coherent across WGPs.
// ---------------------------------------------------------------------------
__device__ __forceinline__ void atomic_max_f32(float* addr, float val) {
    asm volatile("global_atomic_max_num_f32 %0, %1, off scope:SCOPE_DEV"
                 :: "v"(addr), "v"(val)
                 : "memory");
}

// ---------------------------------------------------------------------------
// Kernel 1: hw[r] = h @ W[r]  for r in [0, NUM_RELS], fp32 WMMA.
// grid = (N_NODES/16, NUM_RELS+1), block = 256 (8 waves).
// Wave `w` computes the 16x16 tile at columns [16w, 16w+16) of one 16-row
// strip of one relation. K-loop: 32 x V_WMMA_F32_16X16X4_F32.
//
// A (16x4 f32) lane layout: lanes 0-15: M=lane, K={k,k+1}; lanes 16-31:
// M=lane-16, K={k+2,k+3}  -> one float2 load per lane per step.
// B (4x16 f32) lane layout mirrors A: n = lane&15, same K split.
// C/D (16x16 f32, 8 VGPRs): VGPR v: lanes 0-15 -> M=v, lanes 16-31 -> M=v+8.
// ---------------------------------------------------------------------------
__global__ void rgcn_gemm_kernel(const float* __restrict__ h,
                                 const float* __restrict__ w,
                                 float* __restrict__ hw) {
    const int lane = threadIdx.x & 31;
    const int wave = threadIdx.x >> 5;
    const int m0   = blockIdx.x << 4;          // 16-row strip
    const int r    = blockIdx.y;               // relation
    const int n0   = wave << 4;                // 16-col tile

    const int mrow  = lane & 15;
    const int khalf = (lane >> 4) << 1;        // 0 for lanes 0-15, 2 for 16-31
    const int ncol  = lane & 15;

    const float* __restrict__ hrow = h + (size_t)(m0 + mrow) * N_FEAT;
    const float* __restrict__ Wcol = w + (size_t)r * N_FEAT * N_FEAT + n0 + ncol;

    v8f acc = {};
#pragma unroll 4
    for (int k = 0; k < N_FEAT; k += 4) {
        const float2 av = *(const float2*)(hrow + k + khalf);   // global_load_b64
        v2f a; a.x = av.x; a.y = av.y;
        v2f b;
        b.x = Wcol[(size_t)(k + khalf)     * N_FEAT];
        b.y = Wcol[(size_t)(k + khalf + 1) * N_FEAT];
        acc = __builtin_amdgcn_wmma_f32_16x16x4_f32(
                  /*neg_a=*/false, a, /*neg_b=*/false, b,
                  /*c_mod=*/(short)0, acc, /*reuse_a=*/false, /*reuse_b=*/false);
    }

    float* __restrict__ out = hw + ((size_t)r * N_NODES + m0) * N_FEAT + n0;
    const int mbase = (lane >> 4) << 3;        // 0 or 8
    const int nn    = lane & 15;
#pragma unroll
    for (int v = 0; v < 8; ++v)
        out[(size_t)(mbase + v) * N_FEAT + nn] = acc[v];
}

// ---------------------------------------------------------------------------
// Kernel 2: init agg to -FLT_MAX (sentinel for "no in-edge").
// ---------------------------------------------------------------------------
__global__ void rgcn_init_agg_kernel(float4* __restrict__ agg, int n4) {
    int i = blockIdx.x * blockDim.x + threadIdx.x;
    if (i < n4)
        agg[i] = make_float4(-FLT_MAX, -FLT_MAX, -FLT_MAX, -FLT_MAX);
}

// ---------------------------------------------------------------------------
// Kernel 3: per-edge message + scatter-max. One wave per edge; lane j owns
// the float4 at feature offset 4j. msg = hw[rel, src] + bias[rel];
// agg[dst] = max(agg[dst], msg) via native f32 atomic max.
// ---------------------------------------------------------------------------
__global__ void rgcn_edge_kernel(const float* __restrict__ hw,
                                 const float* __restrict__ bias,
                                 const int* __restrict__ src,
                                 const int* __restrict__ dst,
                                 const int* __restrict__ rel,
                                 float* __restrict__ agg,
                                 int n_edges) {
    const int e    = (int)((blockIdx.x * (size_t)blockDim.x + threadIdx.x) >> 5);
    const int lane = threadIdx.x & 31;
    if (e >= n_edges) return;

    const int s = src[e];
    const int d = dst[e];
    const int r = rel[e];

    const float4 mv = ((const float4*)(hw + ((size_t)r * N_NODES + s) * N_FEAT))[lane];
    const float4 bv = ((const float4*)(bias + (size_t)r * N_FEAT))[lane];

    float* ap = agg + (size_t)d * N_FEAT + (lane << 2);
    atomic_max_f32(ap + 0, mv.x + bv.x);
    atomic_max_f32(ap + 1, mv.y + bv.y);
    atomic_max_f32(ap + 2, mv.z + bv.z);
    atomic_max_f32(ap + 3, mv.w + bv.w);
}

// ---------------------------------------------------------------------------
// Kernel 4: out = relu(hw[self] + (deg>0 ? agg : 0) + bias[self]).
// agg still at -FLT_MAX <=> no in-edges (all real messages are finite).
// ---------------------------------------------------------------------------
__global__ void rgcn_final_kernel(const float4* __restrict__ hw_self,
                                  const float* __restrict__ bias_self,
                                  const float4* __restrict__ agg,
                                  float4* __restrict__ out) {
    int i = blockIdx.x * blockDim.x + threadIdx.x;
    if (i >= N_NODES * N_FEAT / 4) return;

    float4 a = agg[i];
    float4 s = hw_self[i];
    float4 b = *(const float4*)(bias_self + ((i << 2) & (N_FEAT - 1)));

    float ax = (a.x > -FLT_MAX) ? a.x : 0.0f;
    float ay = (a.y > -FLT_MAX) ? a.y : 0.0f;
    float az = (a.z > -FLT_MAX) ? a.z : 0.0f;
    float aw = (a.w > -FLT_MAX) ? a.w : 0.0f;

    float4 o;
    o.x = s.x + ax + b.x; o.x = o.x > 0.0f ? o.x : 0.0f;
    o.y = s.y + ay + b.y; o.y = o.y > 0.0f ? o.y : 0.0f;
    o.z = s.z + az + b.z; o.z = o.z > 0.0f ? o.z : 0.0f;
    o.w = s.w + aw + b.w; o.w = o.w > 0.0f ? o.w : 0.0f;
    out[i] = o;
}

// ---------------------------------------------------------------------------
extern "C" void kernel_launch(void* const* d_in, const int* in_sizes, int n_in,
                              void* d_out, int out_size, void* d_ws, size_t ws_size,
                              hipStream_t stream) {
    const float* h      = (const float*)d_in[0];
    const float* weight = (const float*)d_in[1];
    const float* bias   = (const float*)d_in[2];
    const int*   src    = (const int*)d_in[3];
    const int*   dst    = (const int*)d_in[4];
    const int*   rel    = (const int*)d_in[5];
    float*       out    = (float*)d_out;
    const int n_edges   = in_sizes[3];

    // Workspace layout: hw[(R+1) * N * F] then agg[N * F]  (~256 MB total).
    float* hw  = (float*)d_ws;
    float* agg = hw + (size_t)(NUM_RELS + 1) * N_NODES * N_FEAT;

    // 1) hw[r] = h @ W[r] for all 9 relation slots (incl. self-loop slot).
    dim3 ggrid(N_NODES / 16, NUM_RELS + 1);
    rgcn_gemm_kernel<<<ggrid, 256, 0, stream>>>(h, weight, hw);

    // 2) agg := -FLT_MAX (must re-init every call; ws is stale across replays).
    const int n4 = N_NODES * N_FEAT / 4;
    rgcn_init_agg_kernel<<<(n4 + 255) / 256, 256, 0, stream>>>((float4*)agg, n4);

    // 3) scatter-max messages (1 wave/edge, 8 waves/block).
    const int edge_blocks = (n_edges + 7) / 8;
    rgcn_edge_kernel<<<edge_blocks, 256, 0, stream>>>(hw, bias, src, dst, rel,
                                                      agg, n_edges);

    // 4) self-loop + agg + bias, relu.
    rgcn_final_kernel<<<(n4 + 255) / 256, 256, 0, stream>>>(
        (const float4*)(hw + (size_t)NUM_RELS * N_NODES * N_FEAT),
        bias + (size_t)NUM_RELS * N_FEAT,
        (const float4*)agg,
        (float4*)out);
}